// OxyNet_49529562858436
// MI455X (gfx1250) — compile-verified
//
#include <hip/hip_runtime.h>
#include <math.h>

// ---------------------------------------------------------------------------
// CapsNet forward for MI455X (gfx1250, wave32).
// Heavy GEMMs (primary-caps conv + per-route u_hat) use V_WMMA_F32_16X16X4_F32
// (native fp32 WMMA, 16x16 tiles, K=4). u_hat is never materialized: it is
// recomputed from L2-resident W_route (137.7MB < 192MB L2) in each of the 5
// routing sub-passes, cutting HBM traffic from ~1.8GB to ~150MB.
// ---------------------------------------------------------------------------

typedef __attribute__((ext_vector_type(2))) float v2f;
typedef __attribute__((ext_vector_type(8))) float v8f;

static constexpr int BBATCH = 16;
static constexpr int RR     = 26912;       // routes
static constexpr int NCC    = 10;          // high-level capsules
static constexpr int OCC    = 16;          // high-level capsule dim
static constexpr int H1     = 31;          // conv1 output spatial
static constexpr int H2     = 29;          // pc conv output spatial
static constexpr int S1     = H1 * H1;     // 961
static constexpr int S2     = H2 * H2;     // 841
static constexpr int NBN    = BBATCH * S1; // 15376 elems per channel for BN
static constexpr int UPB    = 256 * S2;    // 215296 floats of u per batch image

// ---------------------------------------------------------------------------
// Kernel 1: conv 3->32 stride2 + bias, and per-channel sum/sumsq for BN.
// grid = (61, 32), block = 256. blockIdx.y = output channel.
// ---------------------------------------------------------------------------
__global__ void conv1_bnstat_kernel(const float* __restrict__ x,
                                    const float* __restrict__ cw,
                                    const float* __restrict__ cb,
                                    float* __restrict__ h,
                                    float* __restrict__ stats) {
    const int oc  = blockIdx.y;
    const int tid = threadIdx.x;
    const int idx = blockIdx.x * blockDim.x + tid;     // over B*S1 = 15376
    float val = 0.0f;
    if (idx < NBN) {
        const int b  = idx / S1;
        const int s2 = idx % S1;
        const int oh = s2 / H1, ow = s2 % H1;
        const float* xp = x + (size_t)b * 3 * 4096 + (2 * oh) * 64 + 2 * ow;
        const float* wp = cw + oc * 27;
        float sum = cb[oc];
#pragma unroll
        for (int ic = 0; ic < 3; ++ic)
#pragma unroll
            for (int kh = 0; kh < 3; ++kh)
#pragma unroll
                for (int kw = 0; kw < 3; ++kw)
                    sum += xp[ic * 4096 + kh * 64 + kw] * wp[ic * 9 + kh * 3 + kw];
        h[(size_t)b * 32 * S1 + (size_t)oc * S1 + s2] = sum;
        val = sum;
    }
    __shared__ float r1[256];
    __shared__ float r2[256];
    r1[tid] = val;
    r2[tid] = val * val;
    __syncthreads();
    for (int s = 128; s > 0; s >>= 1) {
        if (tid < s) { r1[tid] += r1[tid + s]; r2[tid] += r2[tid + s]; }
        __syncthreads();
    }
    if (tid == 0) {
        atomicAdd(&stats[oc], r1[0]);
        atomicAdd(&stats[32 + oc], r2[0]);
    }
}

// ---------------------------------------------------------------------------
// Kernel 2: apply batchnorm (training-mode stats) + ReLU, in place on h.
// ---------------------------------------------------------------------------
__global__ void bn_relu_kernel(float* __restrict__ h,
                               const float* __restrict__ stats,
                               const float* __restrict__ gamma,
                               const float* __restrict__ beta) {
    const int idx = blockIdx.x * blockDim.x + threadIdx.x;
    if (idx >= BBATCH * 32 * S1) return;
    const int oc = (idx / S1) % 32;
    const float invN = 1.0f / (float)NBN;
    const float mean = stats[oc] * invN;
    const float var  = stats[32 + oc] * invN - mean * mean;
    const float v = gamma[oc] * (h[idx] - mean) * rsqrtf(var + 1e-5f) + beta[oc];
    h[idx] = fmaxf(v, 0.0f);
}

// ---------------------------------------------------------------------------
// Kernel 3: PrimaryCaps conv 32->256, 3x3 s1, as implicit GEMM with fp32 WMMA.
// C[oc=256, n=B*841] = W[oc, k=288] x im2col[k, n].  One 16x16 tile per wave.
// A layout (16x4 f32): lanes 0-15 M=lane, VGPR0/1 = K={0,1}; lanes 16-31 K={2,3}.
// B layout (4x16 f32): VGPR0/1 lanes0-15 K={0,1} N=lane; lanes16-31 K={2,3}.
// block = (32,8) -> 8 waves/block; grid = 13456/8 = 1682 (exact, EXEC all-1s).
// ---------------------------------------------------------------------------
__global__ void pcconv_wmma_kernel(const float* __restrict__ h,
                                   const float* __restrict__ pw,
                                   const float* __restrict__ pb,
                                   float* __restrict__ u) {
    const int lm   = threadIdx.x & 15;
    const int half = threadIdx.x >> 4;
    const int wid  = blockIdx.x * blockDim.y + threadIdx.y;
    if (wid >= 16 * S2 * 16 / 16) { /* 13456 waves exactly, never taken */ }
    const int mt = wid & 15;          // oc tile (16 tiles of 16)
    const int nt = wid >> 4;          // spatial tile (841 tiles of 16)
    const int n  = nt * 16 + lm;      // this lane's output column
    const int b    = n / S2;
    const int sidx = n % S2;
    const int oh = sidx / H2, ow = sidx % H2;
    const float* hp = h + (size_t)b * 32 * S1 + oh * H1 + ow;
    const float* ap = pw + (size_t)(mt * 16 + lm) * 288 + 2 * half;

    v8f acc = {0.f, 0.f, 0.f, 0.f, 0.f, 0.f, 0.f, 0.f};
    for (int k0 = 0; k0 < 288; k0 += 4) {
        const v2f a = *(const v2f*)(ap + k0);      // weights: A[m=oc][k]
        const int kk = k0 + 2 * half;
        // im2col gather for k = kk and kk+1
        const int ic0 = kk / 9;
        const int rm0 = kk - ic0 * 9;
        const int kh0 = rm0 / 3, kw0 = rm0 - kh0 * 3;
        const int kk1 = kk + 1;
        const int ic1 = kk1 / 9;
        const int rm1 = kk1 - ic1 * 9;
        const int kh1 = rm1 / 3, kw1 = rm1 - kh1 * 3;
        v2f bb;
        bb.x = hp[ic0 * S1 + kh0 * H1 + kw0];
        bb.y = hp[ic1 * S1 + kh1 * H1 + kw1];
        acc = __builtin_amdgcn_wmma_f32_16x16x4_f32(false, a, false, bb,
                                                    (short)0, acc, false, false);
    }
#pragma unroll
    for (int q = 0; q < 8; ++q) {
        const int oc = mt * 16 + q + 8 * half;      // D: M = q + 8*(lane/16)
        u[(size_t)b * UPB + (size_t)oc * S2 + sidx] = acc[q] + pb[oc];
    }
}

// ---------------------------------------------------------------------------
// Kernel 4: squash over capsule dim (8), in place. u layout: [B, R, 8].
// ---------------------------------------------------------------------------
__global__ void squash_kernel(float* __restrict__ u) {
    const int idx = blockIdx.x * blockDim.x + threadIdx.x;
    if (idx >= BBATCH * RR) return;
    float* p = u + (size_t)idx * 8;
    float t[8];
    float sn = 0.0f;
#pragma unroll
    for (int i = 0; i < 8; ++i) { t[i] = p[i]; sn += t[i] * t[i]; }
    const float sc = sn / ((1.0f + sn) * sqrtf(sn));
#pragma unroll
    for (int i = 0; i < 8; ++i) p[i] = t[i] * sc;
}

// ---------------------------------------------------------------------------
// Per-route u_hat via fp32 WMMA:  C[b=16, co=160] = u_r[16x8] * W_r^T[8x160].
// 10 N-tiles (one per capsule c, since OC=16) x 2 K-steps = 20 WMMAs/route.
// W_route row for route r is 1280 contiguous floats: [co=160][i=8].
// ---------------------------------------------------------------------------
__device__ __forceinline__ void uhat_route(const float* __restrict__ u,
                                           const float* __restrict__ Wr,
                                           int r, int lm, int half,
                                           v8f acc[10]) {
    const float* up = u + (size_t)lm * UPB + (size_t)r * 8;   // A: M=batch=lm
    const float* wp = Wr + (size_t)r * 1280;
    const v2f a0 = *(const v2f*)(up + 2 * half);              // K 0..3
    const v2f a1 = *(const v2f*)(up + 4 + 2 * half);          // K 4..7
#pragma unroll
    for (int t = 0; t < 10; ++t) {
        const float* wb = wp + (t * 16 + lm) * 8 + 2 * half;  // B[k][n=co]
        const v2f b0 = *(const v2f*)(wb);
        const v2f b1 = *(const v2f*)(wb + 4);
        v8f c = {0.f, 0.f, 0.f, 0.f, 0.f, 0.f, 0.f, 0.f};
        c = __builtin_amdgcn_wmma_f32_16x16x4_f32(false, a0, false, b0,
                                                  (short)0, c, false, false);
        c = __builtin_amdgcn_wmma_f32_16x16x4_f32(false, a1, false, b1,
                                                  (short)0, c, false, false);
        acc[t] = c;
    }
}

// ---------------------------------------------------------------------------
// Kernel 5 (s-pass): s[b,c,o] = sum_r c_ij[r,c] * u_hat[b,r,c,o].
// One route per wave, u_hat recomputed (W_route streamed from L2).
// Block-local LDS accumulator (2560 floats) -> one global atomicAdd flush.
// grid = 26912/8 = 3364 exact.
// ---------------------------------------------------------------------------
__global__ void s_pass_kernel(const float* __restrict__ u,
                              const float* __restrict__ Wr,
                              const float* __restrict__ cij,
                              float* __restrict__ sg,
                              int uniform) {
    __shared__ float sblk[2560];
    const int tid = threadIdx.y * 32 + threadIdx.x;
    for (int i = tid; i < 2560; i += 256) sblk[i] = 0.0f;
    __syncthreads();

    const int r = blockIdx.x * 8 + threadIdx.y;
    if (r < RR) {
        const int lm = threadIdx.x & 15;
        const int half = threadIdx.x >> 4;
        v8f acc[10];
        uhat_route(u, Wr, r, lm, half, acc);
#pragma unroll
        for (int t = 0; t < 10; ++t) {
            const float coef = uniform ? (1.0f / (float)RR) : cij[r * 10 + t];
#pragma unroll
            for (int q = 0; q < 8; ++q) {
                // D layout: b = q + 8*(lane/16), o = lane%16, c = tile t
                atomicAdd(&sblk[(q + 8 * half) * 160 + t * 16 + lm],
                          coef * acc[t][q]);
            }
        }
    }
    __syncthreads();
    for (int i = tid; i < 2560; i += 256) atomicAdd(&sg[i], sblk[i]);
}

// ---------------------------------------------------------------------------
// Kernel 6: elementwise squash of s -> v (reference squashes a singleton dim).
// ---------------------------------------------------------------------------
__global__ void squash_v_kernel(const float* __restrict__ s,
                                float* __restrict__ v) {
    const int i = blockIdx.x * blockDim.x + threadIdx.x;
    if (i >= BBATCH * NCC * OCC) return;
    const float x = s[i];
    const float sn = x * x;
    v[i] = sn * x / ((1.0f + sn) * sqrtf(sn));
}

// ---------------------------------------------------------------------------
// Kernel 7 (a-pass): b_ij[r,c] += mean_b sum_o u_hat[b,r,c,o] * v[b,c,o].
// u_hat recomputed; wave-local dot + __shfl_xor tree reduce (wave32);
// single wave owns route r -> plain read-modify-write on b_ij.
// ---------------------------------------------------------------------------
__global__ void a_pass_kernel(const float* __restrict__ u,
                              const float* __restrict__ Wr,
                              const float* __restrict__ vg,
                              float* __restrict__ bij) {
    __shared__ float vsh[2560];
    const int tid = threadIdx.y * 32 + threadIdx.x;
    for (int i = tid; i < 2560; i += 256) vsh[i] = vg[i];
    __syncthreads();

    const int r = blockIdx.x * 8 + threadIdx.y;
    if (r >= RR) return;
    const int lm = threadIdx.x & 15;
    const int half = threadIdx.x >> 4;
    v8f acc[10];
    uhat_route(u, Wr, r, lm, half, acc);
#pragma unroll
    for (int t = 0; t < 10; ++t) {
        float p = 0.0f;
#pragma unroll
        for (int q = 0; q < 8; ++q)
            p += acc[t][q] * vsh[(q + 8 * half) * 160 + t * 16 + lm];
#pragma unroll
        for (int off = 16; off > 0; off >>= 1)
            p += __shfl_xor(p, off, 32);
        if (threadIdx.x == 0)
            bij[r * 10 + t] += p * (1.0f / 16.0f);
    }
}

// ---------------------------------------------------------------------------
// Kernel 8: softmax over routes (axis 0) per capsule c. One block per c.
// ---------------------------------------------------------------------------
__global__ void softmax_routes_kernel(const float* __restrict__ bij,
                                      float* __restrict__ cij) {
    const int c = blockIdx.x;
    const int tid = threadIdx.x;
    __shared__ float sh[256];

    float m = -3.402823466e38f;
    for (int r = tid; r < RR; r += 256) m = fmaxf(m, bij[r * 10 + c]);
    sh[tid] = m;
    __syncthreads();
    for (int s = 128; s > 0; s >>= 1) {
        if (tid < s) sh[tid] = fmaxf(sh[tid], sh[tid + s]);
        __syncthreads();
    }
    m = sh[0];
    __syncthreads();

    float sum = 0.0f;
    for (int r = tid; r < RR; r += 256) sum += expf(bij[r * 10 + c] - m);
    sh[tid] = sum;
    __syncthreads();
    for (int s = 128; s > 0; s >>= 1) {
        if (tid < s) sh[tid] += sh[tid + s];
        __syncthreads();
    }
    const float inv = 1.0f / sh[0];

    for (int r = tid; r < RR; r += 256)
        cij[r * 10 + c] = expf(bij[r * 10 + c] - m) * inv;
}

// ---------------------------------------------------------------------------
// Kernel 9: FC head. out[b,j] = v[b,:160] . fc_w[j,:160] + fc_b[j]. 48 outs.
// ---------------------------------------------------------------------------
__global__ void fc_kernel(const float* __restrict__ v,
                          const float* __restrict__ fw,
                          const float* __restrict__ fb,
                          float* __restrict__ out) {
    const int tid = threadIdx.x;
    if (tid >= 48) return;
    const int b = tid / 3, j = tid % 3;
    float sum = fb[j];
    for (int k = 0; k < 160; ++k) sum += v[b * 160 + k] * fw[j * 160 + k];
    out[b * 3 + j] = sum;
}

// ---------------------------------------------------------------------------
extern "C" void kernel_launch(void* const* d_in, const int* in_sizes, int n_in,
                              void* d_out, int out_size, void* d_ws, size_t ws_size,
                              hipStream_t stream) {
    const float* x    = (const float*)d_in[0];
    const float* cw   = (const float*)d_in[1];
    const float* cb   = (const float*)d_in[2];
    const float* g    = (const float*)d_in[3];
    const float* beta = (const float*)d_in[4];
    const float* pw   = (const float*)d_in[5];
    const float* pb   = (const float*)d_in[6];
    const float* WR   = (const float*)d_in[7];
    const float* fw   = (const float*)d_in[8];
    const float* fb   = (const float*)d_in[9];
    float* out = (float*)d_out;

    // workspace layout (floats): total ~4.48M floats (~17.9 MB)
    float* ws    = (float*)d_ws;
    float* stats = ws;                       // 64
    float* h     = ws + 64;                  // 492032  [16,32,31,31]
    float* u     = h + 492032;               // 3444736 [16,26912,8] (in-place squash)
    float* bij   = u + 3444736;              // 269120  [R,10]
    float* cij   = bij + 269120;             // 269120  [R,10]
    float* sbuf  = cij + 269120;             // 2560    [16,10,16]
    float* vbuf  = sbuf + 2560;              // 2560

    hipMemsetAsync(stats, 0, 64 * sizeof(float), stream);
    hipMemsetAsync(bij, 0, 269120 * sizeof(float), stream);

    conv1_bnstat_kernel<<<dim3(61, 32), 256, 0, stream>>>(x, cw, cb, h, stats);
    bn_relu_kernel<<<(BBATCH * 32 * S1 + 255) / 256, 256, 0, stream>>>(h, stats, g, beta);
    pcconv_wmma_kernel<<<(16 * S2 * 16 / 16) / 8, dim3(32, 8), 0, stream>>>(h, pw, pb, u);
    squash_kernel<<<(BBATCH * RR + 255) / 256, 256, 0, stream>>>(u);

    const int routeBlocks = RR / 8;  // 3364 exact -> full waves, EXEC all-1s
    for (int it = 0; it < 3; ++it) {
        if (it > 0) softmax_routes_kernel<<<NCC, 256, 0, stream>>>(bij, cij);
        hipMemsetAsync(sbuf, 0, 2560 * sizeof(float), stream);
        s_pass_kernel<<<routeBlocks, dim3(32, 8), 0, stream>>>(u, WR, cij, sbuf, it == 0 ? 1 : 0);
        squash_v_kernel<<<NCC, 256, 0, stream>>>(sbuf, vbuf);
        if (it < 2) a_pass_kernel<<<routeBlocks, dim3(32, 8), 0, stream>>>(u, WR, vbuf, bij);
    }
    fc_kernel<<<1, 64, 0, stream>>>(vbuf, fw, fb, out);
}